// MultiLevelPooling_36850819399726
// MI455X (gfx1250) — compile-verified
//
#include <hip/hip_runtime.h>
#include <math.h>

#define NUM_GRAPHS 1024
#define IN_F 256
#define HID 512

typedef __attribute__((ext_vector_type(2))) float v2f;
typedef __attribute__((ext_vector_type(4))) float v4f;
typedef __attribute__((ext_vector_type(8))) float v8f;

// ---------------------------------------------------------------------------
// Kernel 1: segment sum / mean / max pooling.
// batch is sorted, so each graph's rows are contiguous: binary-search bounds.
// One block (256 thr) per graph; non-temporal float4 loads (x is streamed once
// and is bigger than the 192MB L2 -> NT keeps L2 for the reused weights/pools),
// 4-way row parallelism, LDS tree reduce.
// Memory-bound: 205 MB @ 23.3 TB/s ~ 9 us.
// ---------------------------------------------------------------------------
__global__ void __launch_bounds__(256)
pool_kernel(const float* __restrict__ x, const int* __restrict__ batch, int n_nodes,
            float* __restrict__ sum_pool, float* __restrict__ mean_pool,
            float* __restrict__ max_pool) {
  const int g = blockIdx.x;
  __shared__ int s_bounds[2];
  if (threadIdx.x < 2) {
    const int target = g + (int)threadIdx.x;  // lower_bound(batch, g) / (batch, g+1)
    int lo = 0, hi = n_nodes;
    while (lo < hi) {
      int mid = (lo + hi) >> 1;
      if (batch[mid] < target) lo = mid + 1; else hi = mid;
    }
    s_bounds[threadIdx.x] = lo;
  }
  __syncthreads();
  const int start = s_bounds[0];
  const int end   = s_bounds[1];
  const int t  = threadIdx.x;
  const int fq = t & 63;   // which float4 of the 256-wide feature row
  const int rg = t >> 6;   // row group 0..3

  v4f s = {0.f, 0.f, 0.f, 0.f};
  v4f m = {-INFINITY, -INFINITY, -INFINITY, -INFINITY};
  const v4f* x4 = (const v4f*)x;
#pragma unroll 4
  for (int r = start + rg; r < end; r += 4) {
    // Non-temporal: x is a one-pass stream larger than L2.
    v4f v = __builtin_nontemporal_load(&x4[(size_t)r * (IN_F / 4) + fq]);
    s.x += v.x; s.y += v.y; s.z += v.z; s.w += v.w;
    m.x = fmaxf(m.x, v.x); m.y = fmaxf(m.y, v.y);
    m.z = fmaxf(m.z, v.z); m.w = fmaxf(m.w, v.w);
  }
  __shared__ v4f sh_s[256];
  __shared__ v4f sh_m[256];
  sh_s[t] = s; sh_m[t] = m;
  __syncthreads();
  if (t < 64) {
    for (int i = 1; i < 4; ++i) {
      v4f s2 = sh_s[t + 64 * i];
      s.x += s2.x; s.y += s2.y; s.z += s2.z; s.w += s2.w;
      v4f m2 = sh_m[t + 64 * i];
      m.x = fmaxf(m.x, m2.x); m.y = fmaxf(m.y, m2.y);
      m.z = fmaxf(m.z, m2.z); m.w = fmaxf(m.w, m2.w);
    }
    const int cnt = end - start;
    const float inv = 1.0f / fmaxf((float)cnt, 1.0f);
    if (cnt == 0) m = (v4f){0.f, 0.f, 0.f, 0.f};
    v4f mean = {s.x * inv, s.y * inv, s.z * inv, s.w * inv};
    ((v4f*)(sum_pool  + (size_t)g * IN_F))[t] = s;
    ((v4f*)(mean_pool + (size_t)g * IN_F))[t] = mean;
    ((v4f*)(max_pool  + (size_t)g * IN_F))[t] = m;
  }
}

// ---------------------------------------------------------------------------
// Kernel 2/4: C = A[MxK] * B[KxN] + bias, f32, via V_WMMA_F32_16X16X4_F32.
// One wave computes a 16(M) x 64(N) strip: the A fragment (16x4) is reused
// across four 16-wide N tiles. 4 waves per block cover 64 rows of M.
// Fragment layouts per CDNA5 ISA 7.12.2:
//   A 16x4 : lane<16 -> {K=k,k+1}, lane>=16 -> {K=k+2,k+3}; M = lane&15
//   B 4x16 : mirrored (N = lane&15)
//   C 16x16: VGPR v -> M = v + 8*(lane>=16), N = lane&15
// f32 WMMA keeps exact reference precision; GEMM work (~1.1 GFLOP total on
// L2-resident operands) is negligible vs the pooling stream, so no reason
// to downcast to bf16/fp8.
// ---------------------------------------------------------------------------
__global__ void __launch_bounds__(128)
wmma_gemm_bias(const float* __restrict__ A, const float* __restrict__ B,
               const float* __restrict__ bias, float* __restrict__ C,
               int M, int N, int K) {
  const int wave = threadIdx.x >> 5;
  const int lane = threadIdx.x & 31;
  const int half = lane >> 4;
  const int lm   = lane & 15;
  const int m0 = (blockIdx.y * 4 + wave) * 16;
  const int n0 = blockIdx.x * 64;
  if (m0 >= M) return;

  v8f acc0 = {}, acc1 = {}, acc2 = {}, acc3 = {};
  const float* arow = A + (size_t)(m0 + lm) * K + 2 * half;

  for (int k = 0; k < K; k += 4) {
    v2f a;
    a.x = arow[k];
    a.y = arow[k + 1];
    const float* bp = B + (size_t)(k + 2 * half) * N + n0 + lm;
    v2f b0, b1, b2, b3;
    b0.x = bp[0];  b0.y = bp[N + 0];
    b1.x = bp[16]; b1.y = bp[N + 16];
    b2.x = bp[32]; b2.y = bp[N + 32];
    b3.x = bp[48]; b3.y = bp[N + 48];
    acc0 = __builtin_amdgcn_wmma_f32_16x16x4_f32(false, a, false, b0, (short)0, acc0, false, false);
    acc1 = __builtin_amdgcn_wmma_f32_16x16x4_f32(false, a, false, b1, (short)0, acc1, false, false);
    acc2 = __builtin_amdgcn_wmma_f32_16x16x4_f32(false, a, false, b2, (short)0, acc2, false, false);
    acc3 = __builtin_amdgcn_wmma_f32_16x16x4_f32(false, a, false, b3, (short)0, acc3, false, false);
  }

  v8f acc[4] = {acc0, acc1, acc2, acc3};
#pragma unroll
  for (int j = 0; j < 4; ++j) {
    const int n = n0 + 16 * j + lm;
    const float bv = bias[n];
    float* crow = C + (size_t)(m0 + half * 8) * N + n;
#pragma unroll
    for (int v = 0; v < 8; ++v)
      crow[(size_t)v * N] = acc[j][v] + bv;
  }
}

// ---------------------------------------------------------------------------
// Kernel 3: gates (3x 512-dot + sigmoid), softmax over the 3 gates, fusion.
// One block (512 thr) per graph.
// ---------------------------------------------------------------------------
__global__ void __launch_bounds__(512)
gate_fuse_kernel(const float* __restrict__ mean_repr, const float* __restrict__ max_repr,
                 const float* __restrict__ sum_repr,
                 const float* __restrict__ gmw, const float* __restrict__ gmb,
                 const float* __restrict__ gxw, const float* __restrict__ gxb,
                 const float* __restrict__ gsw, const float* __restrict__ gsb,
                 float* __restrict__ pooled) {
  const int g = blockIdx.x;
  const int h = threadIdx.x;
  const size_t idx = (size_t)g * HID + h;
  const float mr = mean_repr[idx];
  const float xr = max_repr[idx];
  const float sr = sum_repr[idx];

  __shared__ float r0[HID], r1[HID], r2[HID];
  __shared__ float w[3];
  r0[h] = mr * gmw[h];
  r1[h] = xr * gxw[h];
  r2[h] = sr * gsw[h];
  __syncthreads();
  for (int off = HID / 2; off >= 1; off >>= 1) {
    if (h < off) {
      r0[h] += r0[h + off];
      r1[h] += r1[h + off];
      r2[h] += r2[h + off];
    }
    __syncthreads();
  }
  if (h == 0) {
    const float gm = 1.f / (1.f + expf(-(r0[0] + gmb[0])));
    const float gx = 1.f / (1.f + expf(-(r1[0] + gxb[0])));
    const float gs = 1.f / (1.f + expf(-(r2[0] + gsb[0])));
    const float mx = fmaxf(gm, fmaxf(gx, gs));
    const float e0 = expf(gm - mx), e1 = expf(gx - mx), e2 = expf(gs - mx);
    const float inv = 1.f / (e0 + e1 + e2);
    w[0] = e0 * inv; w[1] = e1 * inv; w[2] = e2 * inv;
  }
  __syncthreads();
  pooled[idx] = w[0] * mr + w[1] * xr + w[2] * sr;
}

// ---------------------------------------------------------------------------
// Kernel 5: layernorm over the 256-wide feature dim. One block per graph.
// ---------------------------------------------------------------------------
__global__ void __launch_bounds__(256)
layernorm_kernel(const float* __restrict__ e, const float* __restrict__ gamma,
                 const float* __restrict__ beta, float* __restrict__ out) {
  const int g = blockIdx.x;
  const int f = threadIdx.x;
  const float v = e[(size_t)g * IN_F + f];
  __shared__ float red[IN_F];
  __shared__ float stat;
  red[f] = v;
  __syncthreads();
  for (int off = IN_F / 2; off >= 1; off >>= 1) {
    if (f < off) red[f] += red[f + off];
    __syncthreads();
  }
  if (f == 0) stat = red[0] * (1.0f / IN_F);
  __syncthreads();
  const float mu = stat;
  const float d = v - mu;
  red[f] = d * d;
  __syncthreads();
  for (int off = IN_F / 2; off >= 1; off >>= 1) {
    if (f < off) red[f] += red[f + off];
    __syncthreads();
  }
  if (f == 0) stat = red[0] * (1.0f / IN_F);
  __syncthreads();
  const float var = stat;
  out[(size_t)g * IN_F + f] = d * rsqrtf(var + 1e-5f) * gamma[f] + beta[f];
}

// ---------------------------------------------------------------------------
extern "C" void kernel_launch(void* const* d_in, const int* in_sizes, int n_in,
                              void* d_out, int out_size, void* d_ws, size_t ws_size,
                              hipStream_t stream) {
  const float* x      = (const float*)d_in[0];
  const int*   batch  = (const int*)  d_in[1];
  const float* W_mean = (const float*)d_in[2];
  const float* b_mean = (const float*)d_in[3];
  const float* W_max  = (const float*)d_in[4];
  const float* b_max  = (const float*)d_in[5];
  const float* W_sum  = (const float*)d_in[6];
  const float* b_sum  = (const float*)d_in[7];
  const float* g_mean_w = (const float*)d_in[8];
  const float* g_mean_b = (const float*)d_in[9];
  const float* g_max_w  = (const float*)d_in[10];
  const float* g_max_b  = (const float*)d_in[11];
  const float* g_sum_w  = (const float*)d_in[12];
  const float* g_sum_b  = (const float*)d_in[13];
  const float* W_out  = (const float*)d_in[14];
  const float* b_out  = (const float*)d_in[15];
  const float* ln_g   = (const float*)d_in[16];
  const float* ln_b   = (const float*)d_in[17];
  float* out = (float*)d_out;

  const int n_nodes = in_sizes[0] / IN_F;  // 200000

  // Workspace layout (floats)
  float* ws = (float*)d_ws;
  const size_t GF = (size_t)NUM_GRAPHS * IN_F;   // 262144
  const size_t GH = (size_t)NUM_GRAPHS * HID;    // 524288
  float* sum_pool  = ws;
  float* mean_pool = sum_pool + GF;
  float* max_pool  = mean_pool + GF;
  float* mean_repr = max_pool + GF;
  float* max_repr  = mean_repr + GH;
  float* sum_repr  = max_repr + GH;
  float* pooled    = sum_repr + GH;
  float* emb_pre   = pooled + GH;                // total ~12.6 MB

  // 1) segment pooling (memory-bound streaming pass over x, NT loads)
  pool_kernel<<<NUM_GRAPHS, 256, 0, stream>>>(x, batch, n_nodes,
                                              sum_pool, mean_pool, max_pool);

  // 2) three pooled-representation GEMMs: [1024,256]x[256,512] + bias (f32 WMMA)
  {
    dim3 grid(HID / 64, NUM_GRAPHS / 64);
    wmma_gemm_bias<<<grid, 128, 0, stream>>>(mean_pool, W_mean, b_mean, mean_repr,
                                             NUM_GRAPHS, HID, IN_F);
    wmma_gemm_bias<<<grid, 128, 0, stream>>>(max_pool, W_max, b_max, max_repr,
                                             NUM_GRAPHS, HID, IN_F);
    wmma_gemm_bias<<<grid, 128, 0, stream>>>(sum_pool, W_sum, b_sum, sum_repr,
                                             NUM_GRAPHS, HID, IN_F);
  }

  // 3) gated fusion
  gate_fuse_kernel<<<NUM_GRAPHS, HID, 0, stream>>>(mean_repr, max_repr, sum_repr,
                                                   g_mean_w, g_mean_b,
                                                   g_max_w, g_max_b,
                                                   g_sum_w, g_sum_b, pooled);

  // 4) output projection: [1024,512]x[512,256] + bias (f32 WMMA)
  {
    dim3 grid(IN_F / 64, NUM_GRAPHS / 64);
    wmma_gemm_bias<<<grid, 128, 0, stream>>>(pooled, W_out, b_out, emb_pre,
                                             NUM_GRAPHS, IN_F, HID);
  }

  // 5) layernorm -> d_out
  layernorm_kernel<<<NUM_GRAPHS, IN_F, 0, stream>>>(emb_pre, ln_g, ln_b, out);
}